// LorentzXEntropyLoss_60559038874029
// MI455X (gfx1250) — compile-verified
//
#include <hip/hip_runtime.h>

typedef __attribute__((ext_vector_type(2))) float v2f;
typedef __attribute__((ext_vector_type(4))) float v4f;
typedef __attribute__((ext_vector_type(8))) float v8f;

#define NB 8192          // batch rows
#define NC 32000         // classes
#define ND 32            // embedding dim
#define ROWS_PER_WG 16
#define WAVES_PER_WG 4
#define THREADS (WAVES_PER_WG * 32)
#define NTILES (NC / 16) // 2000 column tiles of 16
#define EPS_F 1e-6f

__device__ __forceinline__ float fast_rcp(float x)  { return __builtin_amdgcn_rcpf(x); }
__device__ __forceinline__ float fast_sqrt(float x) { return __builtin_amdgcn_sqrtf(x); }

// One workgroup = 16 batch rows, 4 waves. Each wave sweeps a strided subset of the
// 2000 column tiles with chained V_WMMA_F32_16X16X4_F32, accumulating
// sum_n exp(logit) = sum_n 1/(x + sqrt(x^2-1)) per row (max fixed at 0: logits <= 0).
__global__ __launch_bounds__(THREADS) void lorentz_xent_main(
    const float* __restrict__ pred, const int* __restrict__ tgt,
    const float* __restrict__ emb, float* __restrict__ nll)
{
    __shared__ float s_logit_t[ROWS_PER_WG];
    __shared__ float s_part[WAVES_PER_WG][ROWS_PER_WG];

    const int tid  = threadIdx.x;
    const int wave = tid >> 5;
    const int lane = tid & 31;
    const int half = lane >> 4;   // 0: lanes 0-15 (k 0..15), 1: lanes 16-31 (k 16..31)
    const int l16  = lane & 15;
    const int row0 = blockIdx.x * ROWS_PER_WG;

    // ---- target logit per row (cheap: 32 MACs per row), threads 0..15 ----
    if (tid < ROWS_PER_WG) {
        const int r = row0 + tid;
        const int t = tgt[r];
        const float* pr = pred + (size_t)r * ND;
        const float* er = emb  + (size_t)t * ND;
        float acc = -pr[0] * er[0];                    // Lorentz signature on coord 0
        #pragma unroll
        for (int k = 1; k < ND; ++k) acc = fmaf(pr[k], er[k], acc);
        const float x = fmaxf(-acc, 1.0f + EPS_F);
        s_logit_t[tid] = -logf(x + fast_sqrt(fmaf(x, x, -1.0f)));   // -arccosh(x)
    }

    // ---- A fragment: 16 rows x 32 k, K permuted so each lane loads 16 contiguous floats.
    // WMMA step kk uses k = {half*16 + 2kk, half*16 + 2kk + 1} for this lane's half.
    v2f afrag[8];
    {
        const v4f* ap = (const v4f*)(pred + (size_t)(row0 + l16) * ND + half * 16);
        float ab[16];
        #pragma unroll
        for (int i = 0; i < 4; ++i) {
            const v4f v = ap[i];
            ab[4*i+0] = v.x; ab[4*i+1] = v.y; ab[4*i+2] = v.z; ab[4*i+3] = v.w;
        }
        if (half == 0) ab[0] = -ab[0];                 // fold Lorentz sign into A
        #pragma unroll
        for (int kk = 0; kk < 8; ++kk) { afrag[kk].x = ab[2*kk]; afrag[kk].y = ab[2*kk+1]; }
    }

    float rsum[8];
    #pragma unroll
    for (int j = 0; j < 8; ++j) rsum[j] = 0.0f;

    // ---- sweep column tiles: wave w handles tiles w, w+4, w+8, ... ----
    for (int nt = wave; nt < NTILES; nt += WAVES_PER_WG) {
        const float* bbase = emb + (size_t)(nt * 16 + l16) * ND + half * 16;
        __builtin_prefetch(bbase + (size_t)WAVES_PER_WG * 16 * ND, 0, 1);  // next tile -> global_prefetch_b8

        v2f bfrag[8];
        {
            const v4f* bp = (const v4f*)bbase;
            float bb[16];
            #pragma unroll
            for (int i = 0; i < 4; ++i) {
                const v4f v = bp[i];
                bb[4*i+0] = v.x; bb[4*i+1] = v.y; bb[4*i+2] = v.z; bb[4*i+3] = v.w;
            }
            #pragma unroll
            for (int kk = 0; kk < 8; ++kk) { bfrag[kk].x = bb[2*kk]; bfrag[kk].y = bb[2*kk+1]; }
        }

        v8f c = {};
        #pragma unroll
        for (int kk = 0; kk < 8; ++kk)
            c = __builtin_amdgcn_wmma_f32_16x16x4_f32(false, afrag[kk], false, bfrag[kk],
                                                      (short)0, c, false, false);

        // exp(logit) = exp(-arccosh(x)) = 1/(x + sqrt(x^2 - 1)); no log/exp needed.
        #pragma unroll
        for (int j = 0; j < 8; ++j) {
            const float x = fmaxf(-c[j], 1.0f + EPS_F);
            rsum[j] += fast_rcp(x + fast_sqrt(fmaf(x, x, -1.0f)));
        }
    }

    // ---- reduce N across the 16 lanes of each half (C/D layout: VGPR j -> rows j, j+8) ----
    #pragma unroll
    for (int j = 0; j < 8; ++j) {
        float s = rsum[j];
        s += __shfl_xor(s, 1, 32);
        s += __shfl_xor(s, 2, 32);
        s += __shfl_xor(s, 4, 32);
        s += __shfl_xor(s, 8, 32);
        rsum[j] = s;                                   // valid in lanes 0 and 16
    }
    if (l16 == 0) {
        #pragma unroll
        for (int j = 0; j < 8; ++j) s_part[wave][j + 8 * half] = rsum[j];
    }
    __syncthreads();

    // ---- combine 4 waves (fixed order: deterministic), emit per-row NLL ----
    if (tid < ROWS_PER_WG) {
        const float s = s_part[0][tid] + s_part[1][tid] + s_part[2][tid] + s_part[3][tid];
        nll[row0 + tid] = logf(s) - s_logit_t[tid];    // logZ - logit_target
    }
}

// Deterministic fixed-tree mean over the 8192 per-row NLL values.
__global__ __launch_bounds__(256) void lorentz_xent_reduce(
    const float* __restrict__ nll, float* __restrict__ out)
{
    __shared__ float sh[256];
    float s = 0.0f;
    for (int i = threadIdx.x; i < NB; i += 256) s += nll[i];
    sh[threadIdx.x] = s;
    __syncthreads();
    #pragma unroll
    for (int off = 128; off > 0; off >>= 1) {
        if (threadIdx.x < off) sh[threadIdx.x] += sh[threadIdx.x + off];
        __syncthreads();
    }
    if (threadIdx.x == 0) out[0] = sh[0] * (1.0f / (float)NB);
}

extern "C" void kernel_launch(void* const* d_in, const int* in_sizes, int n_in,
                              void* d_out, int out_size, void* d_ws, size_t ws_size,
                              hipStream_t stream) {
    (void)in_sizes; (void)n_in; (void)out_size; (void)ws_size;
    const float* pred = (const float*)d_in[0];   // [8192, 32] f32
    const int*   tgt  = (const int*)  d_in[1];   // [8192] int32
    const float* emb  = (const float*)d_in[2];   // [32000, 32] f32
    float* nll = (float*)d_ws;                   // B floats of scratch
    float* out = (float*)d_out;                  // scalar mean NLL

    lorentz_xent_main<<<NB / ROWS_PER_WG, THREADS, 0, stream>>>(pred, tgt, emb, nll);
    lorentz_xent_reduce<<<1, 256, 0, stream>>>(nll, out);
}